// HartleySpectralConv_71536975282447
// MI455X (gfx1250) — compile-verified
//
#include <hip/hip_runtime.h>
#include <cmath>
#include <cstdint>

// HartleySpectralConv for MI455X (gfx1250, wave32, WMMA).
// 5 dense fp32 GEMM stages on V_WMMA_F32_16X16X4_F32 + async LDS staging
// via GLOBAL_LOAD_ASYNC_TO_LDS_* drained with s_wait_asynccnt.

typedef __attribute__((ext_vector_type(2))) float v2f;
typedef __attribute__((ext_vector_type(8))) float v8f;

#define PI2_OVER_256 0.0245436926061702596754894014319f  // 2*pi/256

__device__ __forceinline__ v8f wmma_f32_16x16x4(v2f a, v2f b, v8f c) {
  // (neg_a, A, neg_b, B, c_mod, C, reuse_a, reuse_b)
  return __builtin_amdgcn_wmma_f32_16x16x4_f32(false, a, false, b, (short)0, c,
                                               false, false);
}

// ---- async global->LDS staging (GLOBAL_LOAD_ASYNC_TO_LDS_*, ASYNCcnt) -----
#if defined(__gfx1250__) &&                                                   \
    __has_builtin(__builtin_amdgcn_global_load_async_to_lds_b128) &&          \
    __has_builtin(__builtin_amdgcn_global_load_async_to_lds_b32) &&           \
    __has_builtin(__builtin_amdgcn_s_wait_asynccnt)
typedef __attribute__((__vector_size__(4 * sizeof(int)))) int v4i;
typedef __attribute__((address_space(1))) v4i* gp_b128;
typedef __attribute__((address_space(3))) v4i* lp_b128;
typedef __attribute__((address_space(1))) int* gp_b32;
typedef __attribute__((address_space(3))) int* lp_b32;
__device__ __forceinline__ void async_b128(const float* g, float* l) {
  // AS(3) pointer value == low 32 bits of the generic LDS address (ISA 10.2).
  __builtin_amdgcn_global_load_async_to_lds_b128(
      (gp_b128)(uintptr_t)g, (lp_b128)(uint32_t)(uintptr_t)l, 0, 0);
}
__device__ __forceinline__ void async_b32(const float* g, float* l) {
  __builtin_amdgcn_global_load_async_to_lds_b32(
      (gp_b32)(uintptr_t)g, (lp_b32)(uint32_t)(uintptr_t)l, 0, 0);
}
__device__ __forceinline__ void async_wait0() {
  __builtin_amdgcn_s_wait_asynccnt(0);
}
#else
__device__ __forceinline__ void async_b128(const float* g, float* l) {
  l[0] = g[0]; l[1] = g[1]; l[2] = g[2]; l[3] = g[3];
}
__device__ __forceinline__ void async_b32(const float* g, float* l) { *l = *g; }
__device__ __forceinline__ void async_wait0() {}
#endif

// ---- workspace layout (float offsets) -------------------------------------
#define O_PW   0u          // [256][80]   fwd W-stage basis (cols 66..79 zero)
#define O_AH   20480u      // [64][512]   fwd H-stage basis (cos|sin interleaved)
#define O_CHM  53248u      // [256][64]   inverse H-stage CAS basis (transposed)
#define O_CWQ  69632u      // [36][256]   inverse W-stage CAS basis (rows 33..35 zero)
#define O_T    78848u      // [512][512][33]  T[bi][2p+c][v]
#define O_XH   8729600u    // [512][64][33]   xh[bi][u][v]
#define O_OUTF 9810944u    // [512][64][33]   outF[bo][u][v]
#define O_U    10892288u   // [512][256][36]  U[bo][p][v-padded]

// ---- stage 0: build basis matrices (tiny, fully deterministic) ------------
__global__ void basis_kernel(float* __restrict__ ws) {
  float* Pw  = ws + O_PW;
  float* Ah  = ws + O_AH;
  float* Chm = ws + O_CHM;
  float* Cwq = ws + O_CWQ;
  const float invHW  = 1.0f / 65536.0f;                 // rfft2 'forward' norm
  const float invOut = invHW * (1.0f / 33024.0f);       // fft2 'forward' * 1/n
  const int N0 = 256 * 80, N1 = 64 * 512, N2 = 256 * 64, N3 = 36 * 256;
  const int NT = N0 + N1 + N2 + N3;
  for (int idx = blockIdx.x * blockDim.x + threadIdx.x; idx < NT;
       idx += gridDim.x * blockDim.x) {
    if (idx < N0) {
      int q = idx / 80, j = idx % 80;
      float val = 0.0f;
      if (j < 66) {
        int v = j >> 1, c = j & 1;
        int m = (v == 0) ? 0 : (129 - v);               // the mod-129 flipneg
        float bm = PI2_OVER_256 * (float)((m * q) & 255);
        float bv = PI2_OVER_256 * (float)((v * q) & 255);
        val = (c == 0) ? (cosf(bm) + sinf(bv)) * invHW
                       : (cosf(bv) - sinf(bm)) * invHW;
      }
      Pw[idx] = val;
    } else if (idx < N0 + N1) {
      int t = idx - N0;
      int ui = t / 512, k = t % 512, p = k >> 1, c = k & 1;
      int u = (ui < 32) ? ui : (192 + ui);              // rows 0..31, 224..255
      float a = PI2_OVER_256 * (float)((u * p) & 255);
      Ah[t] = (c == 0) ? cosf(a) : sinf(a);
    } else if (idx < N0 + N1 + N2) {
      int t = idx - N0 - N1;
      int p = t / 64, ui = t % 64;
      int u = (ui < 32) ? ui : (192 + ui);
      float a = PI2_OVER_256 * (float)((u * p) & 255);
      Chm[t] = cosf(a) + sinf(a);                       // cas()
    } else {
      int t = idx - N0 - N1 - N2;
      int v = t / 256, q = t % 256;
      float val = 0.0f;
      if (v < 33) {
        float a = PI2_OVER_256 * (float)((v * q) & 255);
        val = (cosf(a) + sinf(a)) * invOut;
      }
      Cwq[t] = val;
    }
  }
}

// ---- stage A: forward along W. GEMM M=131072, K=256, N=66 -----------------
__global__ void __launch_bounds__(160) fwd_w_kernel(const float* __restrict__ x,
                                                    const float* __restrict__ Pw,
                                                    float* __restrict__ T) {
  __shared__ float ax[16 * 256];                        // 16-row x-panel, 16 KB
  const int row0 = blockIdx.x * 16;                     // flat (bi,p) row
  const int tid = threadIdx.x;
  __builtin_prefetch(Pw, 0, 3);                         // global_prefetch_b8
  // contiguous 16 KB tile -> async 16B chunks straight into LDS
  for (int idx = tid; idx < 16 * 64; idx += 160)
    async_b128(&x[(size_t)row0 * 256 + idx * 4], &ax[idx * 4]);
  async_wait0();
  __syncthreads();
  const int wave = tid >> 5, lane = tid & 31;
  const int l16 = lane >> 4, lm = lane & 15;
  const int n = wave * 16 + lm;                         // output col (v,c)
  v8f acc = {};
  for (int k0 = 0; k0 < 256; k0 += 4) {
    const int ka = k0 + 2 * l16;
    v2f a, b;
    a.x = ax[lm * 256 + ka];
    a.y = ax[lm * 256 + ka + 1];
    b.x = Pw[ka * 80 + n];
    b.y = Pw[(ka + 1) * 80 + n];
    acc = wmma_f32_16x16x4(a, b, acc);
  }
  if (n < 66) {
    const int v = n >> 1, c = n & 1;
#pragma unroll
    for (int r = 0; r < 8; ++r) {
      int row = row0 + r + 8 * l16;
      int bi = row >> 8, p = row & 255;
      T[((size_t)bi * 512 + 2 * p + c) * 33 + v] = acc[r];
    }
  }
}

// ---- stage B: forward along H. Per bi: [64x512]x[512x33] ------------------
__global__ void __launch_bounds__(384) fwd_h_kernel(const float* __restrict__ T,
                                                    const float* __restrict__ Ah,
                                                    float* __restrict__ xh) {
  __shared__ float bt[256 * 36];                        // half-K panel of T
  const int bi = blockIdx.x;
  const int tid = threadIdx.x;
  const int wave = tid >> 5, lane = tid & 31;
  const int l16 = lane >> 4, lm = lane & 15;
  const int mt = wave / 3, nt = wave % 3;               // 4 M-tiles x 3 N-tiles
  const float* Tb = T + (size_t)bi * (512 * 33);
  v8f acc = {};
  for (int kh = 0; kh < 2; ++kh) {
    __syncthreads();
    for (int idx = tid; idx < 256 * 36; idx += 384) {
      int r = idx / 36, c = idx % 36;
      if (c < 33) async_b32(&Tb[(size_t)(kh * 256 + r) * 33 + c], &bt[idx]);
      else        bt[idx] = 0.0f;
    }
    async_wait0();
    __syncthreads();
    const float* Arow = Ah + (mt * 16 + lm) * 512 + kh * 256;
    for (int k0 = 0; k0 < 256; k0 += 4) {
      const int ka = k0 + 2 * l16;
      v2f a, b;
      a.x = Arow[ka];
      a.y = Arow[ka + 1];
      b.x = bt[ka * 36 + nt * 16 + lm];
      b.y = bt[(ka + 1) * 36 + nt * 16 + lm];
      acc = wmma_f32_16x16x4(a, b, acc);
    }
  }
  const int v = nt * 16 + lm;
  if (v < 33) {
#pragma unroll
    for (int r = 0; r < 8; ++r) {
      int u = mt * 16 + r + 8 * l16;
      xh[((size_t)bi * 64 + u) * 33 + v] = acc[r];
    }
  }
}

// ---- stage C: channel mixing. out = X1*We + X2*Wo per (u,v) ---------------
__global__ void __launch_bounds__(128) mix_kernel(const float* __restrict__ xh,
                                                  const float* __restrict__ w1,
                                                  const float* __restrict__ w2,
                                                  float* __restrict__ outF) {
  const int pos = blockIdx.x;                           // 64*33 positions
  const int u = pos / 33, v = pos % 33;
  const int rv = (v == 0) ? 0 : (33 - v);
  const float* w = (u < 32) ? w1 : w2;
  const int wrow = (u < 32) ? (32 + u) : (u - 32);
  const int wave = threadIdx.x >> 5, lane = threadIdx.x & 31;
  const int l16 = lane >> 4, lm = lane & 15;
  const int o = wave * 16 + lm;                         // output channel tile
  const float msk = (lm < 8) ? 1.0f : 0.0f;             // B=8 rows, rest zero
  v8f acc = {};
  for (int k0 = 0; k0 < 64; k0 += 4) {
    const int ka = k0 + 2 * l16;
    v2f a1, a2, be, bo;
    {  // A frags: xh[b=lm, i=ka..ka+1] at v and rv
      size_t x0 = ((size_t)((lm & 7) * 64 + ka)) * 2112 + (size_t)u * 33;
      a1.x = msk * xh[x0 + v];   a1.y = msk * xh[x0 + 2112 + v];
      a2.x = msk * xh[x0 + rv];  a2.y = msk * xh[x0 + 2112 + rv];
    }
    {  // B frags: even/odd-symmetrized weights on the fly
      size_t w0a = ((size_t)(ka * 64 + o)) * 2112 + (size_t)wrow * 33;
      size_t w1a = w0a + 64 * 2112;                     // i -> i+1
      float e0 = w[w0a + v], f0 = w[w0a + rv];
      float e1 = w[w1a + v], f1 = w[w1a + rv];
      be.x = 0.5f * (e0 + f0);  bo.x = 0.5f * (e0 - f0);
      be.y = 0.5f * (e1 + f1);  bo.y = 0.5f * (e1 - f1);
    }
    acc = wmma_f32_16x16x4(a1, be, acc);
    acc = wmma_f32_16x16x4(a2, bo, acc);
  }
  if (l16 == 0) {                                       // rows b=0..7 valid
#pragma unroll
    for (int r = 0; r < 8; ++r)
      outF[((size_t)(r * 64 + o)) * 2112 + (size_t)u * 33 + v] = acc[r];
  }
}

// ---- stage D1: inverse along H. Per bo: [256x64]x[64x33] ------------------
__global__ void __launch_bounds__(512) inv_h_kernel(const float* __restrict__ outF,
                                                    const float* __restrict__ Chm,
                                                    float* __restrict__ U) {
  __shared__ float bb[64 * 36];
  const int bo = blockIdx.x;
  const int tid = threadIdx.x;
  for (int idx = tid; idx < 64 * 36; idx += 512) {
    int r = idx / 36, c = idx % 36;
    if (c < 33) async_b32(&outF[(size_t)bo * 2112 + r * 33 + c], &bb[idx]);
    else        bb[idx] = 0.0f;
  }
  async_wait0();
  __syncthreads();
  const int wave = tid >> 5, lane = tid & 31;
  const int l16 = lane >> 4, lm = lane & 15;
  const int mt = wave;                                  // 16 M-tiles of p
  for (int nt = 0; nt < 3; ++nt) {
    v8f acc = {};
    for (int k0 = 0; k0 < 64; k0 += 4) {
      const int ka = k0 + 2 * l16;
      v2f a, b;
      a.x = Chm[(mt * 16 + lm) * 64 + ka];
      a.y = Chm[(mt * 16 + lm) * 64 + ka + 1];
      b.x = bb[ka * 36 + nt * 16 + lm];
      b.y = bb[(ka + 1) * 36 + nt * 16 + lm];
      acc = wmma_f32_16x16x4(a, b, acc);
    }
    const int n = nt * 16 + lm;
    if (n < 36) {
#pragma unroll
      for (int r = 0; r < 8; ++r) {
        int p = mt * 16 + r + 8 * l16;
        U[((size_t)bo * 256 + p) * 36 + n] = acc[r];
      }
    }
  }
}

// ---- stage D2: inverse along W + bias. GEMM M=131072, K=36, N=256 ---------
__global__ void __launch_bounds__(256) inv_w_kernel(const float* __restrict__ U,
                                                    const float* __restrict__ Cwq,
                                                    const float* __restrict__ bias,
                                                    float* __restrict__ y) {
  __shared__ float au[16 * 36];
  const int row0 = blockIdx.x * 16;                     // flat (bo,p) row
  const int tid = threadIdx.x;
  __builtin_prefetch(Cwq, 0, 3);
  // contiguous 2304 B panel (144 x 16 B) -> async b128
  for (int idx = tid; idx < 144; idx += 256)
    async_b128(&U[(size_t)row0 * 36 + idx * 4], &au[idx * 4]);
  async_wait0();
  __syncthreads();
  const int wave = tid >> 5, lane = tid & 31;
  const int l16 = lane >> 4, lm = lane & 15;
  const float bv = bias[(row0 >> 8) & 63];
  for (int half = 0; half < 2; ++half) {
    const int nt = wave * 2 + half;                     // 16 N-tiles
    const int n = nt * 16 + lm;
    v8f acc = {};
    for (int k0 = 0; k0 < 36; k0 += 4) {
      const int ka = k0 + 2 * l16;
      v2f a, b;
      a.x = au[lm * 36 + ka];
      a.y = au[lm * 36 + ka + 1];
      b.x = Cwq[ka * 256 + n];
      b.y = Cwq[(ka + 1) * 256 + n];
      acc = wmma_f32_16x16x4(a, b, acc);
    }
#pragma unroll
    for (int r = 0; r < 8; ++r) {
      int m = r + 8 * l16;
      y[(size_t)(row0 + m) * 256 + n] = acc[r] + bv;
    }
  }
}

extern "C" void kernel_launch(void* const* d_in, const int* in_sizes, int n_in,
                              void* d_out, int out_size, void* d_ws, size_t ws_size,
                              hipStream_t stream) {
  (void)in_sizes; (void)n_in; (void)out_size; (void)ws_size;
  const float* x  = (const float*)d_in[0];
  const float* w1 = (const float*)d_in[1];
  const float* w2 = (const float*)d_in[2];
  const float* b  = (const float*)d_in[3];
  float* ws   = (float*)d_ws;
  float* Pw   = ws + O_PW;
  float* Ah   = ws + O_AH;
  float* Chm  = ws + O_CHM;
  float* Cwq  = ws + O_CWQ;
  float* T    = ws + O_T;
  float* xh   = ws + O_XH;
  float* outF = ws + O_OUTF;
  float* U    = ws + O_U;
  float* y    = (float*)d_out;

  basis_kernel<<<308, 256, 0, stream>>>(ws);
  fwd_w_kernel<<<8192, 160, 0, stream>>>(x, Pw, T);
  fwd_h_kernel<<<512, 384, 0, stream>>>(T, Ah, xh);
  mix_kernel<<<2112, 128, 0, stream>>>(xh, w1, w2, outF);
  inv_h_kernel<<<512, 512, 0, stream>>>(outF, Chm, U);
  inv_w_kernel<<<8192, 256, 0, stream>>>(U, Cwq, b, y);
}